// DoubleGCN_1864015806551
// MI455X (gfx1250) — compile-verified
//
#include <hip/hip_runtime.h>
#include <math.h>

// ---------------------------------------------------------------------------
// DoubleGCN on MI455X (gfx1250). Memory-bound on edge gather/scatter
// (~4.4 GB @ 23.3 TB/s ~ 190us); dense GEMMs (15 GFLOP total) done with
// V_WMMA_F32_16X16X4_F32 at full fp32 precision (GEMM cost is negligible, so
// there is no reason to trade precision for matrix-unit rate here).
// ---------------------------------------------------------------------------

#define NNODES 50000
#define NEDGES 800000
#define NREL   4
#define FEAT   256
#define HID    128
#define OUTD   32
#define NBATCH 10000
#define BNEPS  1e-5f

typedef __attribute__((ext_vector_type(2))) float v2f;
typedef __attribute__((ext_vector_type(8))) float v8f;

// ---------------------------------------------------------------------------
// WMMA fp32 GEMM: C[M, NC] = X[M, K] @ W[K, NC]
// grid.x = M/16 row tiles, grid.y = NC/(NT*16) col strips, block = 1 wave(32).
// Per k-step of 4: A frag reused across NT column tiles.
// f32 16x16x4 layouts (ISA 7.12.2):
//   A: lane L -> row L%16, a[j] = A[row][2*(L/16)+j]
//   B: lane L -> col L%16, b[j] = B[2*(L/16)+j][col]
//   C: reg j  -> C[j + 8*(L/16)][L%16]
// ---------------------------------------------------------------------------
template <int NT>
__global__ __launch_bounds__(32) void gemm_wmma_f32(const float* __restrict__ X,
                                                    const float* __restrict__ W,
                                                    float* __restrict__ C,
                                                    int K, int NC) {
  const int lane = threadIdx.x;
  const int m16  = lane & 15;
  const int half = lane >> 4;  // 0 or 1
  const int rowBase = blockIdx.x * 16;
  const int colBase = blockIdx.y * (NT * 16);

  v8f acc[NT] = {};

  const float* xrow = X + (size_t)(rowBase + m16) * K;
  for (int k = 0; k < K; k += 4) {
    const int kb = k + 2 * half;
    v2f a;
    a.x = xrow[kb];
    a.y = xrow[kb + 1];
    const float* w0 = W + (size_t)kb * NC + colBase + m16;
#pragma unroll
    for (int t = 0; t < NT; ++t) {
      v2f b;
      b.x = w0[t * 16];       // row kb
      b.y = w0[t * 16 + NC];  // row kb+1
      acc[t] = __builtin_amdgcn_wmma_f32_16x16x4_f32(
          /*neg_a=*/false, a, /*neg_b=*/false, b,
          /*c_mod=*/(short)0, acc[t], /*reuse_a=*/false, /*reuse_b=*/false);
    }
  }

  float* cp = C + (size_t)(rowBase + 8 * half) * NC + colBase + m16;
#pragma unroll
  for (int t = 0; t < NT; ++t) {
#pragma unroll
    for (int j = 0; j < 8; ++j) {
      cp[(size_t)j * NC + t * 16] = acc[t][j];
    }
  }
}

// ---------------------------------------------------------------------------
// Degree / normalization
// ---------------------------------------------------------------------------
__global__ void set_ones_f32(float* __restrict__ p, int n) {
  int i = blockIdx.x * blockDim.x + threadIdx.x;
  if (i < n) p[i] = 1.0f;  // self-loop contributes 1 to every degree
}

__global__ void zero_f32(float* __restrict__ p, int n) {
  int i = blockIdx.x * blockDim.x + threadIdx.x;
  if (i < n) p[i] = 0.0f;
}

__global__ void deg_count(const int* __restrict__ dst, float* __restrict__ deg, int nE) {
  int e = blockIdx.x * blockDim.x + threadIdx.x;
  if (e < nE) atomicAdd(&deg[dst[e]], 1.0f);
}

__global__ void inplace_rsqrt(float* __restrict__ p, int n) {
  int i = blockIdx.x * blockDim.x + threadIdx.x;
  if (i < n) p[i] = rsqrtf(p[i]);  // deg >= 1 always (self loops)
}

// ---------------------------------------------------------------------------
// Aggregation: out[n,:] starts as bias + dinv[n]^2 * h[n,:] (self-loop term),
// then edge scatter adds dinv[s]*dinv[d] * h[s,:] into out[d,:].
// ---------------------------------------------------------------------------
__global__ void agg_init(const float* __restrict__ h, const float* __restrict__ dinv,
                         const float* __restrict__ bias, float* __restrict__ out,
                         int total4, int Dq) {
  int i = blockIdx.x * blockDim.x + threadIdx.x;
  if (i >= total4) return;
  int n  = i / Dq;
  int cg = i % Dq;
  float w = dinv[n];
  w *= w;
  float4 hv = ((const float4*)h)[i];
  float4 b  = ((const float4*)bias)[cg];
  float4 o;
  o.x = w * hv.x + b.x;
  o.y = w * hv.y + b.y;
  o.z = w * hv.z + b.z;
  o.w = w * hv.w + b.w;
  ((float4*)out)[i] = o;
}

// One wave per edge; 32 lanes x float4 = one full 128-float row.
__global__ void scatter_d128(const float* __restrict__ h, const float* __restrict__ dinv,
                             const int* __restrict__ src, const int* __restrict__ dst,
                             float* __restrict__ out, int nE) {
  int t = blockIdx.x * blockDim.x + threadIdx.x;
  int e = t >> 5;
  if (e >= nE) return;
  int lane = t & 31;
  int s = src[e];
  int d = dst[e];
  float w = dinv[s] * dinv[d];
  float4 hv = ((const float4*)(h + (size_t)s * 128))[lane];
  float* o = out + (size_t)d * 128 + lane * 4;
  atomicAdd(o + 0, w * hv.x);
  atomicAdd(o + 1, w * hv.y);
  atomicAdd(o + 2, w * hv.z);
  atomicAdd(o + 3, w * hv.w);
}

// 8 threads per edge; 8 x float4 = one full 32-float row.
__global__ void scatter_d32(const float* __restrict__ h, const float* __restrict__ dinv,
                            const int* __restrict__ src, const int* __restrict__ dst,
                            float* __restrict__ out, int nE) {
  int t = blockIdx.x * blockDim.x + threadIdx.x;
  int e = t >> 3;
  if (e >= nE) return;
  int g = t & 7;
  int s = src[e];
  int d = dst[e];
  float w = dinv[s] * dinv[d];
  float4 hv = ((const float4*)(h + (size_t)s * 32))[g];
  float* o = out + (size_t)d * 32 + g * 4;
  atomicAdd(o + 0, w * hv.x);
  atomicAdd(o + 1, w * hv.y);
  atomicAdd(o + 2, w * hv.z);
  atomicAdd(o + 3, w * hv.w);
}

// ---------------------------------------------------------------------------
// BatchNorm (training stats over node axis, biased variance) + ReLU
// ---------------------------------------------------------------------------
__global__ void bn_stats(const float* __restrict__ x, float* __restrict__ sums,
                         int nRows, int D) {
  // blockDim.x == 256, D divides 256 (D = 128 or 32)
  int c   = threadIdx.x % D;
  int rpb = blockDim.x / D;
  int r0  = blockIdx.x * rpb + threadIdx.x / D;
  int stride = gridDim.x * rpb;
  float s = 0.0f, ss = 0.0f;
  for (int r = r0; r < nRows; r += stride) {
    float v = x[(size_t)r * D + c];
    s += v;
    ss += v * v;
  }
  atomicAdd(&sums[c], s);
  atomicAdd(&sums[D + c], ss);
}

__global__ void bn_finalize(const float* __restrict__ sums, const float* __restrict__ g,
                            const float* __restrict__ be, float* __restrict__ scsh,
                            int D, float invN) {
  int c = threadIdx.x;
  if (c >= D) return;
  float mean = sums[c] * invN;
  float var  = sums[D + c] * invN - mean * mean;
  float rstd = rsqrtf(var + BNEPS);
  float sc = g[c] * rstd;
  scsh[c]     = sc;
  scsh[D + c] = be[c] - sc * mean;
}

__global__ void bn_apply_relu(float* __restrict__ x, const float* __restrict__ scsh,
                              int total4, int Dq) {
  int i = blockIdx.x * blockDim.x + threadIdx.x;
  if (i >= total4) return;
  int c = (i % Dq) * 4;
  int D = Dq * 4;
  float4 v = ((const float4*)x)[i];
  v.x = fmaxf(0.0f, v.x * scsh[c + 0] + scsh[D + c + 0]);
  v.y = fmaxf(0.0f, v.y * scsh[c + 1] + scsh[D + c + 1]);
  v.z = fmaxf(0.0f, v.z * scsh[c + 2] + scsh[D + c + 2]);
  v.w = fmaxf(0.0f, v.w * scsh[c + 3] + scsh[D + c + 3]);
  ((float4*)x)[i] = v;
}

// ---------------------------------------------------------------------------
// Gather batch nodes + log_softmax over O=32 channels. One wave32 per node:
// lane == channel, wave-wide shuffle max/sum reductions.
// ---------------------------------------------------------------------------
__global__ void logsoftmax_gather(const float* __restrict__ x2, const int* __restrict__ batch,
                                  float* __restrict__ out, int rel, int nB) {
  int t = blockIdx.x * blockDim.x + threadIdx.x;
  int wid = t >> 5;
  if (wid >= nB) return;
  int lane = t & 31;
  int node = batch[wid];
  float v = x2[(size_t)node * OUTD + lane];
  float m = v;
#pragma unroll
  for (int off = 16; off > 0; off >>= 1) m = fmaxf(m, __shfl_xor(m, off, 32));
  float ex = __expf(v - m);
  float s = ex;
#pragma unroll
  for (int off = 16; off > 0; off >>= 1) s += __shfl_xor(s, off, 32);
  out[(size_t)wid * (NREL * OUTD) + rel * OUTD + lane] = v - m - __logf(s);
}

// ---------------------------------------------------------------------------
// Host launcher
// ---------------------------------------------------------------------------
static inline int cdiv(long long a, long long b) { return (int)((a + b - 1) / b); }

extern "C" void kernel_launch(void* const* d_in, const int* in_sizes, int n_in,
                              void* d_out, int out_size, void* d_ws, size_t ws_size,
                              hipStream_t stream) {
  (void)in_sizes; (void)n_in; (void)out_size; (void)ws_size;

  const float* features = (const float*)d_in[0];
  const int*   eidx     = (const int*)d_in[1];   // [R][2][E]
  const int*   batch    = (const int*)d_in[2];   // [B]
  const float* W1a      = (const float*)d_in[3]; // [R][F][H]
  const float* b1a      = (const float*)d_in[4]; // [R][H]
  const float* g1a      = (const float*)d_in[5];
  const float* be1a     = (const float*)d_in[6];
  const float* W2a      = (const float*)d_in[7]; // [R][H][O]
  const float* b2a      = (const float*)d_in[8]; // [R][O]
  const float* g2a      = (const float*)d_in[9];
  const float* be2a     = (const float*)d_in[10];
  float* out = (float*)d_out;                    // [B][R*O]

  // Workspace partition (reused across relations): ~64.3 MB
  char* ws = (char*)d_ws;
  size_t off = 0;
  auto take = [&](size_t bytes) -> float* {
    float* p = (float*)(ws + off);
    off = (off + bytes + 255) & ~(size_t)255;
    return p;
  };
  float* dinv = take((size_t)NNODES * 4);
  float* h1   = take((size_t)NNODES * HID * 4);
  float* x1   = take((size_t)NNODES * HID * 4);
  float* h2   = take((size_t)NNODES * OUTD * 4);
  float* x2   = take((size_t)NNODES * OUTD * 4);
  float* sums = take(2 * HID * 4);
  float* scsh = take(2 * HID * 4);

  const int TB = 256;

  for (int r = 0; r < NREL; ++r) {
    const int* src = eidx + (size_t)r * 2 * NEDGES;
    const int* dst = src + NEDGES;
    const float* W1r  = W1a  + (size_t)r * FEAT * HID;
    const float* b1r  = b1a  + (size_t)r * HID;
    const float* g1r  = g1a  + (size_t)r * HID;
    const float* be1r = be1a + (size_t)r * HID;
    const float* W2r  = W2a  + (size_t)r * HID * OUTD;
    const float* b2r  = b2a  + (size_t)r * OUTD;
    const float* g2r  = g2a  + (size_t)r * OUTD;
    const float* be2r = be2a + (size_t)r * OUTD;

    // --- degrees -> dinv -------------------------------------------------
    set_ones_f32<<<cdiv(NNODES, TB), TB, 0, stream>>>(dinv, NNODES);
    deg_count<<<cdiv(NEDGES, TB), TB, 0, stream>>>(dst, dinv, NEDGES);
    inplace_rsqrt<<<cdiv(NNODES, TB), TB, 0, stream>>>(dinv, NNODES);

    // --- layer 1: h1 = X @ W1r (WMMA fp32) -------------------------------
    gemm_wmma_f32<4><<<dim3(NNODES / 16, HID / 64), 32, 0, stream>>>(
        features, W1r, h1, FEAT, HID);

    // --- aggregate: x1 = b1 + selfloop + edge scatter --------------------
    {
      int total4 = NNODES * HID / 4;
      agg_init<<<cdiv(total4, TB), TB, 0, stream>>>(h1, dinv, b1r, x1, total4, HID / 4);
      scatter_d128<<<cdiv((long long)NEDGES * 32, TB), TB, 0, stream>>>(
          h1, dinv, src, dst, x1, NEDGES);
      // --- BN1 + ReLU ----------------------------------------------------
      zero_f32<<<1, 2 * HID, 0, stream>>>(sums, 2 * HID);
      bn_stats<<<256, TB, 0, stream>>>(x1, sums, NNODES, HID);
      bn_finalize<<<1, HID, 0, stream>>>(sums, g1r, be1r, scsh, HID, 1.0f / NNODES);
      bn_apply_relu<<<cdiv(total4, TB), TB, 0, stream>>>(x1, scsh, total4, HID / 4);
    }

    // --- layer 2: h2 = x1 @ W2r (WMMA fp32) ------------------------------
    gemm_wmma_f32<2><<<dim3(NNODES / 16, 1), 32, 0, stream>>>(x1, W2r, h2, HID, OUTD);

    {
      int total4 = NNODES * OUTD / 4;
      agg_init<<<cdiv(total4, TB), TB, 0, stream>>>(h2, dinv, b2r, x2, total4, OUTD / 4);
      scatter_d32<<<cdiv((long long)NEDGES * 8, TB), TB, 0, stream>>>(
          h2, dinv, src, dst, x2, NEDGES);
      // --- BN2 + ReLU ----------------------------------------------------
      zero_f32<<<1, 2 * OUTD, 0, stream>>>(sums, 2 * OUTD);
      bn_stats<<<256, TB, 0, stream>>>(x2, sums, NNODES, OUTD);
      bn_finalize<<<1, OUTD, 0, stream>>>(sums, g2r, be2r, scsh, OUTD, 1.0f / NNODES);
      bn_apply_relu<<<cdiv(total4, TB), TB, 0, stream>>>(x2, scsh, total4, OUTD / 4);
    }

    // --- gather batch nodes + log_softmax --------------------------------
    logsoftmax_gather<<<cdiv((long long)NBATCH * 32, TB), TB, 0, stream>>>(
        x2, batch, out, r, NBATCH);
  }
}